// MoCo_73443940762097
// MI455X (gfx1250) — compile-verified
//
#include <hip/hip_runtime.h>
#include <hip/hip_bf16.h>

// ---------------------------------------------------------------------------
// MoCo grouped-queue logits for MI455X (gfx1250, wave32, WMMA + TDM)
//   q = l2norm(im_q @ W_q); k = l2norm(im_k @ W_k)
//   l_pos[n] = q[n].k[n];  l_neg[n,:] = q[n,:] @ queues[(label[n]-1)%4]
//   logits = [l_pos, l_neg] / 0.07 ; labels = zeros
// Dominant GEMM: v_wmma_f32_16x16x4_f32 with the B tile (queues) streamed
// into LDS by the Tensor Data Mover (tensor_load_to_lds, TENSORcnt).
// ---------------------------------------------------------------------------

typedef float v2f __attribute__((ext_vector_type(2)));
typedef float v8f __attribute__((ext_vector_type(8)));
typedef unsigned int u32x4 __attribute__((ext_vector_type(4)));
typedef int i32x4 __attribute__((ext_vector_type(4)));
typedef int i32x8 __attribute__((ext_vector_type(8)));

#define N_SAMP 512
#define D_IN   2048
#define DIM    128
#define KQ     65536
#define N_GRP  4
#define OUT_LD 65537            // 1 + KQ
#define INV_T  (1.0f / 0.07f)

#define A_LDS_STRIDE 132        // (4*row) banks -> conflict-free column reads
#define B_LDS_STRIDE 72         // 64 data dwords + 8 pad dwords (TDM pad)

// ---------------------------------------------------------------------------
// 1) Raw projection: out[m,c] = sum_d im[m,d] * W[d,c]   (32-row x 128-col tile)
// ---------------------------------------------------------------------------
__global__ __launch_bounds__(128)
void proj_kernel(const float* __restrict__ im, const float* __restrict__ W,
                 float* __restrict__ outraw)
{
    const int m0 = blockIdx.x * 32;
    const int c  = threadIdx.x;                 // 0..127
    __shared__ float ims[32][64];
    float acc[32];
#pragma unroll
    for (int m = 0; m < 32; ++m) acc[m] = 0.0f;

    for (int d0 = 0; d0 < D_IN; d0 += 64) {
        for (int e = c; e < 32 * 64; e += 128) {
            int r = e >> 6, dd = e & 63;
            ims[r][dd] = im[(size_t)(m0 + r) * D_IN + d0 + dd];
        }
        __syncthreads();
        for (int dd = 0; dd < 64; ++dd) {
            float w = W[(size_t)(d0 + dd) * DIM + c];
#pragma unroll
            for (int m = 0; m < 32; ++m)
                acc[m] = fmaf(ims[m][dd], w, acc[m]);
        }
        __syncthreads();
    }
#pragma unroll
    for (int m = 0; m < 32; ++m)
        outraw[(size_t)(m0 + m) * DIM + c] = acc[m];
}

// ---------------------------------------------------------------------------
// 2) Norms, normalized q, and l_pos -> logits column 0
// ---------------------------------------------------------------------------
__global__ __launch_bounds__(128)
void finalize_kernel(const float* __restrict__ qraw, const float* __restrict__ kraw,
                     float* __restrict__ qn, float* __restrict__ out)
{
    const int n = blockIdx.x;
    const int c = threadIdx.x;
    float qv = qraw[n * DIM + c];
    float kv = kraw[n * DIM + c];
    __shared__ float red[3][128];
    red[0][c] = qv * qv;
    red[1][c] = kv * kv;
    red[2][c] = qv * kv;
    __syncthreads();
    for (int s = 64; s > 0; s >>= 1) {
        if (c < s) {
            red[0][c] += red[0][c + s];
            red[1][c] += red[1][c + s];
            red[2][c] += red[2][c + s];
        }
        __syncthreads();
    }
    float nq = fmaxf(sqrtf(red[0][0]), 1e-12f);
    float nk = fmaxf(sqrtf(red[1][0]), 1e-12f);
    qn[n * DIM + c] = qv / nq;
    if (c == 0)
        out[(size_t)n * OUT_LD] = (red[2][0] / (nq * nk)) * INV_T;
}

// ---------------------------------------------------------------------------
// 3) Route samples into per-queue index lists
// ---------------------------------------------------------------------------
__global__ __launch_bounds__(512)
void route_kernel(const int* __restrict__ label, int* __restrict__ idx,
                  int* __restrict__ cnt)
{
    __shared__ int lcnt[N_GRP];
    const int t = threadIdx.x;
    if (t < N_GRP) lcnt[t] = 0;
    __syncthreads();
    int r = ((label[t] - 1) % N_GRP + N_GRP) % N_GRP;
    int pos = atomicAdd(&lcnt[r], 1);
    idx[r * N_SAMP + pos] = t;
    __syncthreads();
    if (t < N_GRP) cnt[t] = lcnt[t];
}

// ---------------------------------------------------------------------------
// 4) labels output (512 int32 zeros; bit pattern == 0.0f)
// ---------------------------------------------------------------------------
__global__ __launch_bounds__(512)
void zero_labels_kernel(float* __restrict__ out)
{
    out[(size_t)N_SAMP * OUT_LD + threadIdx.x] = 0.0f;
}

// ---------------------------------------------------------------------------
// 5) Main grouped GEMM: l_neg via v_wmma_f32_16x16x4_f32 + TDM B staging
//    Block = 128 threads (4 waves). Tile: M=64 (wave w -> rows w*16..+15),
//    N=64 (4 WMMA n-subtiles per wave). K = 128 in steps of 4.
//    A (gathered normalized-q rows) staged in LDS, stride 132 floats.
//    B (queues tile 128x64, row stride 65536) pulled into LDS by a single
//    tensor_load_to_lds; D# pad fields insert 8 dwords per 64-dword row so
//    the LDS row stride is 72 floats -> the two 16-lane halves of a B
//    fragment hit disjoint bank ranges.
//    WMMA f32 layouts (ISA 7.12.2):
//      A 16x4 : lanes 0-15 carry K={0,1}, lanes 16-31 carry K={2,3}
//      B 4x16 : v0 = rows K=0(l0-15)/K=2(l16-31), v1 = rows K=1/K=3
//      C/D    : vgpr j -> M=j (lanes 0-15), M=j+8 (lanes 16-31)
// ---------------------------------------------------------------------------
__global__ __launch_bounds__(128)
void lneg_wmma_kernel(const float* __restrict__ qn,      // [512][128]
                      const float* __restrict__ queues,  // [4][128][65536]
                      const int* __restrict__ idx,       // [4][512]
                      const int* __restrict__ cnt,       // [4]
                      float* __restrict__ out)           // [512][65537]
{
    const int g     = blockIdx.z;
    const int count = cnt[g];
    const int m0    = blockIdx.y * 64;
    if (m0 >= count) return;                       // empty M-tile for this group
    const int n0 = blockIdx.x * 64;

    __shared__ float Alds[64][A_LDS_STRIDE];       // 33792 B
    __shared__ float Blds[128][B_LDS_STRIDE];      // 36864 B (TDM-padded rows)

    const int tid = threadIdx.x;

    // --- Kick off the TDM transfer of the B tile: queues[g][0:128][n0:n0+64]
    if (tid < 32) {
        unsigned long long ga =
            (unsigned long long)(const void*)(queues + (size_t)g * DIM * KQ + n0);
        u32x4 g0 = {};
        g0[0] = 1u;                                        // count=1, user D#
        g0[1] = (unsigned)(unsigned long long)(&Blds[0][0]); // LDS byte offset
        g0[2] = (unsigned)(ga & 0xFFFFFFFFull);
        g0[3] = (unsigned)((ga >> 32) & 0x1FFFFFFull) | (2u << 30); // type=2
        i32x8 g1 = {};
        // data_size=2 (4B) | pad_enable | pad_interval=5 (64 dw) | pad_amount=7 (8 dw)
        g1[0] = (2 << 16) | (1 << 20) | (5 << 22) | (7 << 25);
        g1[1] = 0;                                         // tensor_dim0 lo16 = 0
        g1[2] = 1 | (DIM << 16);       // tensor_dim0 hi16 (65536) | tensor_dim1 lo16
        g1[3] = (64 << 16);            // tensor_dim1 hi16 = 0 | tile_dim0 = 64
        g1[4] = DIM;                   // tile_dim1 = 128 | tile_dim2 = 0
        g1[5] = KQ;                    // tensor_dim0_stride[31:0] = 65536
        g1[6] = 0;
        g1[7] = 0;
        i32x4 z4 = {};
#if __clang_major__ >= 23
        i32x8 z8 = {};
        __builtin_amdgcn_tensor_load_to_lds(g0, g1, z4, z4, z8, 0);
#else
        __builtin_amdgcn_tensor_load_to_lds(g0, g1, z4, z4, 0);
#endif
    }

    // --- Gather A tile in parallel with the TDM (zero-pad rows beyond count)
    for (int e = tid; e < 64 * DIM; e += 128) {
        int r = e >> 7, c = e & (DIM - 1);
        int m = m0 + r;
        float v = 0.0f;
        if (m < count) {
            int s = idx[g * N_SAMP + m];
            v = qn[(size_t)s * DIM + c];
        }
        Alds[r][c] = v;
    }
    if (tid < 32)
        __builtin_amdgcn_s_wait_tensorcnt(0);      // wave0 owns TENSORcnt
    __syncthreads();                               // A + B visible to all waves

    const int wave = tid >> 5;                     // M sub-tile 0..3
    const int lane = tid & 31;
    const int lm   = lane & 15;                    // column / row-in-half index
    const int kh   = (lane >> 4) << 1;             // 0 for lanes 0-15, 2 for 16-31

    v8f acc0 = {}, acc1 = {}, acc2 = {}, acc3 = {};

    for (int kk = 0; kk < DIM; kk += 4) {
        // A fragment (this wave's 16 rows)
        v2f a;
        a.x = Alds[wave * 16 + lm][kk + kh];
        a.y = Alds[wave * 16 + lm][kk + kh + 1];

        // B fragments for 4 N-subtiles from the TDM-staged LDS tile
        const int r0 = kk + kh;
        v2f b0; b0.x = Blds[r0][lm];      b0.y = Blds[r0 + 1][lm];
        v2f b1; b1.x = Blds[r0][lm + 16]; b1.y = Blds[r0 + 1][lm + 16];
        v2f b2; b2.x = Blds[r0][lm + 32]; b2.y = Blds[r0 + 1][lm + 32];
        v2f b3; b3.x = Blds[r0][lm + 48]; b3.y = Blds[r0 + 1][lm + 48];

        acc0 = __builtin_amdgcn_wmma_f32_16x16x4_f32(false, a, false, b0,
                                                     (short)0, acc0, false, false);
        acc1 = __builtin_amdgcn_wmma_f32_16x16x4_f32(false, a, false, b1,
                                                     (short)0, acc1, false, false);
        acc2 = __builtin_amdgcn_wmma_f32_16x16x4_f32(false, a, false, b2,
                                                     (short)0, acc2, false, false);
        acc3 = __builtin_amdgcn_wmma_f32_16x16x4_f32(false, a, false, b3,
                                                     (short)0, acc3, false, false);
    }

    // Scatter D back to logits rows (skip padded rows), fusing the 1/T scale.
    const int mbase = m0 + wave * 16 + (lane >> 4) * 8;  // M = j (+8 for hi half)
#pragma unroll
    for (int j = 0; j < 8; ++j) {
        int m = mbase + j;
        if (m < count) {
            int s = idx[g * N_SAMP + m];
            float* orow = out + (size_t)s * OUT_LD + 1 + n0 + lm;
            orow[0]  = acc0[j] * INV_T;
            orow[16] = acc1[j] * INV_T;
            orow[32] = acc2[j] * INV_T;
            orow[48] = acc3[j] * INV_T;
        }
    }
}

// ---------------------------------------------------------------------------
// Launcher
// ---------------------------------------------------------------------------
extern "C" void kernel_launch(void* const* d_in, const int* in_sizes, int n_in,
                              void* d_out, int out_size, void* d_ws, size_t ws_size,
                              hipStream_t stream)
{
    (void)in_sizes; (void)n_in; (void)out_size; (void)ws_size;

    const float* im_q   = (const float*)d_in[0];
    const float* im_k   = (const float*)d_in[1];
    const float* W_q    = (const float*)d_in[2];
    const float* W_k    = (const float*)d_in[3];
    const float* queues = (const float*)d_in[4];
    const int*   label  = (const int*)d_in[5];
    float*       out    = (float*)d_out;

    // Workspace layout (floats): qraw | kraw | qn | idx(int) | cnt(int)
    float* ws   = (float*)d_ws;
    float* qraw = ws;                         // 512*128
    float* kraw = ws + N_SAMP * DIM;          // 512*128
    float* qn   = ws + 2 * N_SAMP * DIM;      // 512*128
    int*   idx  = (int*)(ws + 3 * N_SAMP * DIM);   // 4*512
    int*   cnt  = (int*)(ws + 3 * N_SAMP * DIM + N_GRP * N_SAMP);

    proj_kernel<<<N_SAMP / 32, 128, 0, stream>>>(im_q, W_q, qraw);
    proj_kernel<<<N_SAMP / 32, 128, 0, stream>>>(im_k, W_k, kraw);
    finalize_kernel<<<N_SAMP, 128, 0, stream>>>(qraw, kraw, qn, out);
    route_kernel<<<1, N_SAMP, 0, stream>>>(label, idx, cnt);
    zero_labels_kernel<<<1, N_SAMP, 0, stream>>>(out);

    dim3 grid(KQ / 64, N_SAMP / 64, N_GRP);   // (1024, 8, 4)
    lneg_wmma_kernel<<<grid, 128, 0, stream>>>(qn, queues, idx, cnt, out);
}